// GAF_47175920779575
// MI455X (gfx1250) — compile-verified
//
#include <hip/hip_runtime.h>
#include <hip/hip_bf16.h>

typedef __attribute__((ext_vector_type(2))) float v2f;
typedef __attribute__((ext_vector_type(8))) float v8f;

#define GAF_B 16
#define GAF_L 2048
#define GAF_EPS 1e-8f

// ---------------------------------------------------------------------------
// Kernel 1: per-row min/max, then s = clip(2*(x-min)/(max-min+eps)-1) and
// r = sqrt(1 - s^2). One block per row. (16 rows x 2048 -> trivial cost.)
// ---------------------------------------------------------------------------
__global__ void gaf_stats_kernel(const float* __restrict__ x,
                                 float* __restrict__ s,
                                 float* __restrict__ r) {
    __shared__ float smin[256];
    __shared__ float smax[256];

    const int row = blockIdx.x;          // 0..15
    const int tid = threadIdx.x;         // 0..255
    const float* xr = x + row * GAF_L;

    float mn = 3.402823466e+38f;
    float mx = -3.402823466e+38f;
    for (int l = tid; l < GAF_L; l += 256) {
        float v = xr[l];
        mn = fminf(mn, v);
        mx = fmaxf(mx, v);
    }
    smin[tid] = mn;
    smax[tid] = mx;
    __syncthreads();
    for (int off = 128; off > 0; off >>= 1) {
        if (tid < off) {
            smin[tid] = fminf(smin[tid], smin[tid + off]);
            smax[tid] = fmaxf(smax[tid], smax[tid + off]);
        }
        __syncthreads();
    }
    const float minv = smin[0];
    const float maxv = smax[0];
    const float inv  = 2.0f / (maxv - minv + GAF_EPS);

    for (int l = tid; l < GAF_L; l += 256) {
        float sc = (xr[l] - minv) * inv - 1.0f;
        sc = fminf(fmaxf(sc, -1.0f + GAF_EPS), 1.0f - GAF_EPS);
        s[row * GAF_L + l] = sc;
        r[row * GAF_L + l] = sqrtf(fmaxf(0.0f, 1.0f - sc * sc));
    }
}

// ---------------------------------------------------------------------------
// Kernel 2: gaf[b,i,j] = s_i*s_j - r_i*r_j via V_WMMA_F32_16X16X4_F32.
// One wave handles a 16x256 output strip: A built once, 16 WMMA tiles,
// all loads/stores at immediate offsets from per-wave base pointers.
// A (16x4 f32): lanes 0-15 VGPR0=K0=s_i, VGPR1=K1=r_i; lanes 16-31 K2=K3=0.
// B (4x16 f32): VGPR0 = {K0 row (s_j) | K2 row (0)},
//               VGPR1 = {K1 row (-r_j) | K3 row (0)}.
// D (16x16 f32): lane 0-15 -> column, VGPR v -> row v + 8*(lane>=16).
// ---------------------------------------------------------------------------
__global__ void gaf_wmma_kernel(const float* __restrict__ s,
                                const float* __restrict__ r,
                                float* __restrict__ out) {
    const int lane   = threadIdx.x & 31;
    const int lane16 = lane & 15;
    const int hi     = lane >> 4;
    const float fmask = hi ? 0.0f : 1.0f;   // one cndmask, reused all tiles

    // global wave id: 16 batches * 128 i-tiles * 8 j-chunks = 16384 waves
    const unsigned w = blockIdx.x * 8u + (threadIdx.x >> 5);
    const int jchunk = (int)(w & 7u) * 256;        // column base of this strip
    const int i0     = (int)((w >> 3) & 127u) * 16;
    const int b      = (int)(w >> 10);

    const float* sb = s + b * GAF_L;
    const float* rb = r + b * GAF_L;

    // A matrix: unconditional loads (address valid in all lanes), then mask.
    const float asv = sb[i0 + lane16];
    const float arv = rb[i0 + lane16];
    v2f a;
    a.x = asv * fmask;   // K=0 (s_i) in lanes 0-15, 0 in lanes 16-31
    a.y = arv * fmask;   // K=1 (r_i)

    // Per-wave base pointers; everything below is immediate offsets.
    const float* sj = sb + jchunk + lane16;
    const float* rj = rb + jchunk + lane16;
    float* op = out + ((size_t)b * GAF_L + (size_t)(i0 + 8 * hi)) * GAF_L
                    + (size_t)(jchunk + lane16);

#pragma unroll
    for (int jt = 0; jt < 16; ++jt) {
        const int jo = jt * 16;              // compile-time element offset

        const float sv = sj[jo];             // global_load_b32 base+imm
        const float rv = rj[jo];

        v2f bm;
        bm.x = sv * fmask;                   // K=0 row of B (s_j)
        bm.y = -(rv * fmask);                // K=1 row of B (-r_j)

        v8f c = {};
        c = __builtin_amdgcn_wmma_f32_16x16x4_f32(
                /*neg_a=*/false, a,
                /*neg_b=*/false, bm,
                /*c_mod=*/(short)0, c,
                /*reuse_a=*/false, /*reuse_b=*/false);

#pragma unroll
        for (int v = 0; v < 8; ++v) {
            // row m = v + 8*hi, col = jchunk + jo + lane16
            op[(size_t)(v * GAF_L + jo)] = c[v];   // global_store_b32 base+imm
        }
    }
}

// ---------------------------------------------------------------------------
extern "C" void kernel_launch(void* const* d_in, const int* in_sizes, int n_in,
                              void* d_out, int out_size, void* d_ws, size_t ws_size,
                              hipStream_t stream) {
    (void)in_sizes; (void)n_in; (void)out_size; (void)ws_size;

    const float* x = (const float*)d_in[0];
    float* out = (float*)d_out;

    float* sbuf = (float*)d_ws;                 // 16*2048 floats
    float* rbuf = sbuf + GAF_B * GAF_L;         // 16*2048 floats (ws = 256 KB)

    gaf_stats_kernel<<<GAF_B, 256, 0, stream>>>(x, sbuf, rbuf);

    // 16384 waves / 8 waves-per-block = 2048 blocks of 256 threads
    gaf_wmma_kernel<<<2048, 256, 0, stream>>>(sbuf, rbuf, out);
}